// LatentBlock_35622458753598
// MI455X (gfx1250) — compile-verified
//
#include <hip/hip_runtime.h>
#include <math.h>

typedef float v2f __attribute__((ext_vector_type(2)));
typedef float v8f __attribute__((ext_vector_type(8)));
typedef unsigned int u32x4 __attribute__((ext_vector_type(4)));
typedef int i32x4 __attribute__((ext_vector_type(4)));
typedef int i32x8 __attribute__((ext_vector_type(8)));

#define HWDIM 64
#define PLANE 4096  // 64*64

#if defined(__has_builtin)
#if __has_builtin(__builtin_amdgcn_tensor_load_to_lds) && \
    __has_builtin(__builtin_amdgcn_s_wait_tensorcnt)
#define USE_TDM 1
#endif
#endif
#ifndef USE_TDM
#define USE_TDM 0
#endif

static __device__ __forceinline__ v8f wmma4(v2f a, v2f b, v8f c) {
  return __builtin_amdgcn_wmma_f32_16x16x4_f32(false, a, false, b, (short)0, c,
                                               false, false);
}

// Implicit-GEMM conv via V_WMMA_F32_16X16X4_F32.
// Block = 8 waves sharing one 32-out-channel slab; each wave owns 32 pixels
// (2x2 WMMA register blocking). Per (ky,kx) the 32co x CINW weight slab is
// staged into LDS by the Tensor Data Mover (double-buffered, overlapped with
// compute); A operands then come from ds_load_b64, B from coalesced global
// loads with a zero-page fallback pointer for SAME padding (no predicated
// loads -> EXEC all-ones for WMMA).
// MODE 0: out = conv+bias | MODE 1: sigmoid | MODE 2: tanh + GRU update (+out2)
template <int K, int MODE, int COUT, int CINW>
__global__ __launch_bounds__(256) void conv_wmma_k(
    const float* __restrict__ xin, long xinBStride,
    const float* __restrict__ hin, long hinBStride,
    const float* __restrict__ wt, const float* __restrict__ bias,
    const float* __restrict__ zpad, float* __restrict__ out,
    const float* __restrict__ zbuf,  // (B,128,PLANE), MODE 2
    const float* __restrict__ hold,  // (B,64,PLANE) or null, MODE 2
    float* __restrict__ out2, long out2BStride) {
  constexpr int KK = K * K;
  constexpr int PAD = K / 2;
  constexpr unsigned COT = COUT / 32;

  __shared__ float asl[2][32 * CINW];  // double-buffered weight slab

  const unsigned tid = threadIdx.x;
  const unsigned lane = tid & 31;
  const unsigned wv = tid >> 5;  // 0..7
  const unsigned blk = blockIdx.x;

  // block -> (image, co-slab, row-group); wave -> pixel block (all exact)
  const unsigned b = blk / (COT * 16);
  unsigned rem = blk % (COT * 16);
  const unsigned coT = rem >> 4;
  const unsigned pb = (rem & 15) * 8 + wv;  // 0..127
  const unsigned y = pb >> 1;
  const unsigned x0 = (pb & 1) << 5;

  const unsigned hf = lane >> 4;  // K pair {0,1}/{2,3} per ISA layouts
  const unsigned n = lane & 15;   // M index for A, pixel index for B
  const unsigned coBase = coT * 32;
  const bool issuer = (wv == 0);

  v8f acc[2][2] = {{{}, {}}, {{}, {}}};

#if USE_TDM
  // Stage s = ky*K+kx: 2D tile, rows = 32 co (stride KK*CINW), cols = CINW.
  auto issueStage = [&](unsigned s) {
    const float* gptr = wt + ((size_t)coBase * KK + s) * CINW;
    const unsigned long long ga = (unsigned long long)(const void*)gptr;
    const unsigned ldsOff =
        (unsigned)(unsigned long long)(const void*)&asl[s & 1][0];
    u32x4 g0;
    g0.x = 1u;  // count=1, user descriptor, no gather
    g0.y = ldsOff;
    g0.z = (unsigned)(ga & 0xffffffffull);
    g0.w = (unsigned)(ga >> 32) | (2u << 30);  // type=2 ("image")
    const unsigned td0 = CINW, td1 = 32u, tile0 = CINW, tile1 = 32u;
    const unsigned long long st0 = (unsigned long long)KK * CINW;
    i32x8 g1;
    g1[0] = (int)(2u << 16);  // data_size = 4B, workgroup_mask = 0
    g1[1] = (int)((td0 & 0xffffu) << 16);
    g1[2] = (int)((td0 >> 16) | ((td1 & 0xffffu) << 16));
    g1[3] = (int)((td1 >> 16) | (tile0 << 16));
    g1[4] = (int)tile1;  // tile_dim2 = 0
    g1[5] = (int)(st0 & 0xffffffffull);
    g1[6] = (int)((st0 >> 32) & 0xffffull);  // dim1 stride unused (2D)
    g1[7] = 0;
    const i32x4 z4 = {0, 0, 0, 0};
    const i32x8 z8 = {0, 0, 0, 0, 0, 0, 0, 0};
    // 6-arg toolchain variant: (g0, g1, g2, g3, ext, cpol)
    __builtin_amdgcn_tensor_load_to_lds(g0, g1, z4, z4, z8, 0);
  };
  if (issuer) issueStage(0);
#endif

#pragma unroll 1
  for (unsigned s = 0; s < (unsigned)KK; ++s) {
#if USE_TDM
    if (issuer) {
      if (s + 1 < (unsigned)KK) {
        issueStage(s + 1);
        __builtin_amdgcn_s_wait_tensorcnt(1);  // stage s complete
      } else {
        __builtin_amdgcn_s_wait_tensorcnt(0);
      }
    }
#else
    {  // cooperative fallback fill of the weight slab
      const float* gp = wt + ((size_t)coBase * KK + s) * CINW;
      float* dst = &asl[s & 1][0];
      for (unsigned i = tid; i < 32u * (CINW / 2); i += 256u) {
        const unsigned r = i / (CINW / 2), c = (i % (CINW / 2)) * 2;
        *(v2f*)&dst[r * CINW + c] =
            *(const v2f*)&gp[(size_t)r * KK * CINW + c];
      }
    }
#endif
    __syncthreads();  // slab s visible to all waves

    const unsigned ky = s / K, kx = s % K;
    const int yy = (int)y + (int)ky - PAD;
    const bool rowOK = (yy >= 0) && (yy < HWDIM);
    const int xxA = (int)(x0 + n) + (int)kx - PAD;
    const int xxB = xxA + 16;
    const bool okA = rowOK && (xxA >= 0) && (xxA < HWDIM);
    const bool okB = rowOK && (xxB >= 0) && (xxB < HWDIM);
    const float* lb0 = &asl[s & 1][n * CINW + 2 * hf];
    const float* lb1 = &asl[s & 1][(16 + n) * CINW + 2 * hf];

    for (int part = 0; part < 2; ++part) {
      const float* src;
      long bs;
      unsigned wOff;
      if (part == 0) {
        if (!xin) continue;
        src = xin; bs = xinBStride; wOff = 0;
      } else {
        if (!hin) continue;
        src = hin; bs = hinBStride; wOff = (CINW > 64) ? 64u : 0u;
      }
      const float* rowPtr = src + (size_t)b * bs + (long)yy * HWDIM;
      const float* pBA = okA ? (rowPtr + xxA) : zpad;  // hoisted cndmask
      const float* pBB = okB ? (rowPtr + xxB) : zpad;
#pragma unroll
      for (unsigned ci = 0; ci < 64; ci += 4) {
        v2f a0 = *(const v2f*)(lb0 + wOff + ci);  // ds_load_b64
        v2f a1 = *(const v2f*)(lb1 + wOff + ci);
        const long c0 = (long)(ci + 2 * hf) * PLANE;
        v2f b0, b1;
        b0.x = pBA[c0];
        b0.y = pBA[c0 + PLANE];
        b1.x = pBB[c0];
        b1.y = pBB[c0 + PLANE];
        acc[0][0] = wmma4(a0, b0, acc[0][0]);
        acc[0][1] = wmma4(a0, b1, acc[0][1]);
        acc[1][0] = wmma4(a1, b0, acc[1][0]);
        acc[1][1] = wmma4(a1, b1, acc[1][1]);
      }
    }
    __syncthreads();  // all waves done with slab s before buffer reuse
  }

  // C/D layout: VGPR j -> M = j + 8*hf, N = lane&15
  const size_t rowOff = (size_t)y * HWDIM;
  const unsigned px0 = x0 + n;
#pragma unroll
  for (int ct = 0; ct < 2; ++ct) {
#pragma unroll
    for (int pt = 0; pt < 2; ++pt) {
      const size_t po = rowOff + px0 + 16 * pt;
#pragma unroll
      for (int j = 0; j < 8; ++j) {
        const unsigned co = coBase + 16 * ct + j + 8 * hf;
        float v = acc[ct][pt][j] + bias[co];
        const size_t oidx = ((size_t)b * COUT + co) * PLANE + po;
        if (MODE == 0) {
          out[oidx] = v;
        } else if (MODE == 1) {
          out[oidx] = 1.f / (1.f + __expf(-v));
        } else {
          const float c = tanhf(v);
          const float z = zbuf[((size_t)b * 128 + 64 + co) * PLANE + po];
          const float ho =
              hold ? hold[((size_t)b * 64 + co) * PLANE + po] : 0.f;
          const float hn = (1.f - z) * ho + z * c;
          out[oidx] = hn;
          if (out2) out2[(size_t)b * out2BStride + (size_t)co * PLANE + po] = hn;
        }
      }
    }
  }
}

// w (Cout,Cin,K,K) -> wt (Cout,K,K,Cin): ci-innermost rows for TDM/b64 fetch
__global__ __launch_bounds__(256) void wtrans_k(const float* __restrict__ w,
                                                float* __restrict__ wt,
                                                unsigned Cin, unsigned KK,
                                                unsigned total) {
  unsigned i = blockIdx.x * blockDim.x + threadIdx.x;
  if (i >= total) return;
  unsigned kk = i % KK;
  unsigned t = i / KK;
  unsigned ci = t % Cin;
  unsigned co = t / Cin;
  wt[((size_t)co * KK + kk) * Cin + ci] = w[i];
}

// r*h in place over the r-half of the gates buffer
__global__ __launch_bounds__(256) void rh_mul_k(float* __restrict__ gates,
                                                const float* __restrict__ h,
                                                unsigned total) {
  unsigned i = blockIdx.x * blockDim.x + threadIdx.x;
  if (i >= total) return;
  unsigned b = i / (64 * PLANE);
  unsigned c = i % (64 * PLANE);
  size_t g = (size_t)b * (128 * PLANE) + c;
  gates[g] = gates[g] * h[i];
}

__global__ __launch_bounds__(256) void zero_k(float* __restrict__ p,
                                              unsigned total) {
  unsigned i = blockIdx.x * blockDim.x + threadIdx.x;
  if (i < total) p[i] = 0.f;
}

template <int K>
static void run_branch(const float* x, const float* wg, const float* bg,
                       const float* wc, const float* bc, float* gates,
                       float* hbuf, const float* zpad, float* wtg, float* wtc,
                       float* outB, hipStream_t stream) {
  const long FRAME = 64L * PLANE;
  {
    unsigned tg = 128 * 128 * K * K;
    wtrans_k<<<dim3((tg + 255) / 256), dim3(256), 0, stream>>>(wg, wtg, 128,
                                                               K * K, tg);
    unsigned tc = 64 * 128 * K * K;
    wtrans_k<<<dim3((tc + 255) / 256), dim3(256), 0, stream>>>(wc, wtc, 128,
                                                               K * K, tc);
  }
  for (int t = 0; t < 6; ++t) {  // seq_len=3 real frames + n_ahead=3 zero
    const float* xt = (t < 3) ? (x + (size_t)t * FRAME) : nullptr;
    const float* hprev = (t > 0) ? hbuf : nullptr;
    {
      conv_wmma_k<K, 1, 128, 128>
          <<<dim3(16 * 4 * 16), dim3(256), 0, stream>>>(
              xt, 3 * FRAME, hprev, FRAME, wtg, bg, zpad, gates, nullptr,
              nullptr, nullptr, 0L);
    }
    const float* rhp = nullptr;
    if (hprev) {
      unsigned total = 16 * 64 * PLANE;
      rh_mul_k<<<dim3((total + 255) / 256), dim3(256), 0, stream>>>(gates, hbuf,
                                                                    total);
      rhp = gates;  // channels 0..63 now hold r*h, per-b stride 128*PLANE
    }
    {
      float* o2 = (t >= 3) ? (outB + (size_t)(t - 3) * FRAME) : nullptr;
      conv_wmma_k<K, 2, 64, 128>
          <<<dim3(16 * 2 * 16), dim3(256), 0, stream>>>(
              xt, 3 * FRAME, rhp, 128L * PLANE, wtc, bc, zpad, hbuf, gates,
              hprev, o2, 3 * FRAME);
    }
  }
}

extern "C" void kernel_launch(void* const* d_in, const int* in_sizes, int n_in,
                              void* d_out, int out_size, void* d_ws,
                              size_t ws_size, hipStream_t stream) {
  const float* x = (const float*)d_in[0];
  const float* w1 = (const float*)d_in[1];
  const float* b1 = (const float*)d_in[2];
  // dict order: x, w1x1, b1x1, seq_len, n_ahead, then (wg,bg,wc,bc) k=3,5,7
  const float* WG[3] = {(const float*)d_in[5], (const float*)d_in[9],
                        (const float*)d_in[13]};
  const float* BG[3] = {(const float*)d_in[6], (const float*)d_in[10],
                        (const float*)d_in[14]};
  const float* WC[3] = {(const float*)d_in[7], (const float*)d_in[11],
                        (const float*)d_in[15]};
  const float* BC[3] = {(const float*)d_in[8], (const float*)d_in[12],
                        (const float*)d_in[16]};

  const long FRAME = 64L * PLANE;
  float* out0 = (float*)d_out;  // x0: (48,64,64,64)
  float* outBr[3] = {out0 + 48 * FRAME, out0 + 2 * 48 * FRAME,
                     out0 + 3 * 48 * FRAME};

  // workspace layout (floats)
  float* gates = (float*)d_ws;              // 16*128*PLANE = 8388608
  float* hbuf = gates + 16L * 128 * PLANE;  // 16*64*PLANE  = 4194304
  float* zpad = hbuf + 16L * 64 * PLANE;    // 64*PLANE     = 262144 zeros
  float* wtg = zpad + 64L * PLANE;          // up to 128*128*49 = 802816
  float* wtc = wtg + 128L * 128 * 49;       // up to 64*128*49  = 401408

  {  // zero scratch pad used as OOB target (deterministic each call)
    unsigned total = 64 * PLANE;
    zero_k<<<dim3((total + 255) / 256), dim3(256), 0, stream>>>(zpad, total);
  }

  {  // 1x1 projection over all 48 frames; layout already ci-innermost (KK=1)
    conv_wmma_k<1, 0, 64, 64><<<dim3(48 * 2 * 16), dim3(256), 0, stream>>>(
        x, FRAME, nullptr, 0L, w1, b1, zpad, out0, nullptr, nullptr, nullptr,
        0L);
  }

  run_branch<3>(x, WG[0], BG[0], WC[0], BC[0], gates, hbuf, zpad, wtg, wtc,
                outBr[0], stream);
  run_branch<5>(x, WG[1], BG[1], WC[1], BC[1], gates, hbuf, zpad, wtg, wtc,
                outBr[1], stream);
  run_branch<7>(x, WG[2], BG[2], WC[2], BC[2], gates, hbuf, zpad, wtg, wtc,
                outBr[2], stream);
}